// BrainConstructor_4097398800874
// MI455X (gfx1250) — compile-verified
//
#include <hip/hip_runtime.h>
#include <hip/hip_bf16.h>

// ---------------------------------------------------------------------------
// BrainConstructor for MI455X (gfx1250, wave32).
//   Phase 1 : logits for all N^2 pairs (VALU gelu-dot), stored to ws (16.8MB,
//             L2-resident on the 192MB L2).
//   Top-K   : 2-level 64K-bin radix select on order-preserving uint of the
//             float logit -> exact 32-bit threshold -> atomic compaction.
//   Sort    : single-workgroup bitonic sort of 32768 8B keys in 256KB of the
//             320KB LDS -> reference (top_k) descending order, deterministic.
//   Phase 2 : probs = sigmoid(stored logit, recomputed per ref formula from
//             hs/htb); edge features via V_WMMA_F32_16X16X4_F32 (fp32 WMMA).
// ---------------------------------------------------------------------------

typedef float v2f __attribute__((ext_vector_type(2)));
typedef float v8f __attribute__((ext_vector_type(8)));

#define HIST_BINS 65536
#define SORTN     32768   // >= K (20961), power of 2

__device__ __forceinline__ float gelu_f(float x) {
    // exact gelu: x * 0.5 * (1 + erf(x / sqrt(2)))
    return 0.5f * x * (1.0f + erff(x * 0.70710678118654752440f));
}

__device__ __forceinline__ unsigned ordered_of(float f) {
    unsigned u = __float_as_uint(f);
    return (u & 0x80000000u) ? ~u : (u | 0x80000000u);  // ascending order map
}

// ---------------------------------------------------------------- zero scratch
__global__ void k_zero(unsigned* p, unsigned n) {
    unsigned i = blockIdx.x * blockDim.x + threadIdx.x;
    if (i < n) p[i] = 0u;
}

// ------------------------------------------------- hs = x@W1[:D], htb = x@W1[D:]+b1
// One wave computes one 16x16 fp32 output tile with V_WMMA_F32_16X16X4_F32.
__global__ void k_gemm_hsht(const float* __restrict__ x, const float* __restrict__ W1,
                            const float* __restrict__ b1,
                            float* __restrict__ hs, float* __restrict__ htb,
                            int N, int D) {
    const int m0 = blockIdx.x * 16, n0 = blockIdx.y * 16, mat = blockIdx.z;
    const int L = threadIdx.x, mr = L & 15, hi = L >> 4;
    const float* W = W1 + (size_t)mat * D * D;   // row-major [2D][D]
    v8f c = {};
    for (int k0 = 0; k0 < D; k0 += 4) {
        const int ka = k0 + hi * 2;
        v2f a, b;
        a.x = x[(size_t)(m0 + mr) * D + ka];
        a.y = x[(size_t)(m0 + mr) * D + ka + 1];
        b.x = W[(size_t)ka * D + n0 + mr];
        b.y = W[(size_t)(ka + 1) * D + n0 + mr];
        c = __builtin_amdgcn_wmma_f32_16x16x4_f32(false, a, false, b, (short)0, c,
                                                  false, false);
    }
    float* out = (mat == 0) ? hs : htb;
    const float badd = (mat == 1) ? b1[n0 + mr] : 0.0f;
    for (int r = 0; r < 8; ++r) {
        const int M = r + hi * 8;   // C/D layout: VGPR r -> M=r (lanes 0-15), M=r+8
        out[(size_t)(m0 + M) * D + n0 + mr] = c[r] + badd;
    }
}

// ------------------------------- Phase-1 logits + 64K-bin histogram (hi 16 bits)
__global__ void k_logits_hist(const float* __restrict__ hs, const float* __restrict__ htb,
                              const float* __restrict__ W2, const float* __restrict__ b2,
                              float* __restrict__ logits, unsigned* __restrict__ hist1,
                              int N, int D) {
    __shared__ float hsT[64 * 64];
    __shared__ float htT[64 * 64];
    __shared__ float w2s[64];
    const int i0 = blockIdx.x * 64, j0 = blockIdx.y * 64;
    const int t = threadIdx.x;  // 0..255
    for (int idx = t; idx < 64 * D; idx += 256) {
        hsT[idx] = hs[(size_t)(i0 + idx / D) * D + (idx % D)];
        htT[idx] = htb[(size_t)(j0 + idx / D) * D + (idx % D)];
    }
    if (t < D) w2s[t] = W2[t];
    __syncthreads();
    const float b2v = b2[0];
    for (int pp = 0; pp < 16; ++pp) {
        const int p = t * 16 + pp, i = p >> 6, j = p & 63;
        const float* hr = &hsT[i * 64];
        const float* tr = &htT[j * 64];
        float acc = 0.0f;
        #pragma unroll 8
        for (int d = 0; d < 64; ++d) acc = fmaf(gelu_f(hr[d] + tr[d]), w2s[d], acc);
        float lg = acc + b2v;
        if (i0 + i == j0 + j) lg = -__builtin_inff();   // mask diagonal
        logits[(size_t)(i0 + i) * N + (j0 + j)] = lg;
        atomicAdd(&hist1[ordered_of(lg) >> 16], 1u);
    }
}

// ---------------------------------------------- histogram scan (top-down) phases
// phase 0: hist1, target=K      -> ctrl[0]=T, ctrl[1]=n_gt, ctrl[2]=rem
// phase 1: hist2, target=ctrl[2]-> ctrl[3]=L, ctrl[4]=n_gt2, ctrl[5]=rem2
__global__ void k_scan(const unsigned* __restrict__ hist, unsigned* ctrl,
                       int phase, unsigned Ktarget) {
    __shared__ unsigned part[1024];
    __shared__ unsigned pre[1024];
    const int t = threadIdx.x;
    const unsigned target = (phase == 0) ? Ktarget : ctrl[2];
    const int base = HIST_BINS - 64 * (t + 1);   // thread t owns descending chunk
    unsigned s = 0;
    for (int b = base + 63; b >= base; --b) s += hist[b];
    part[t] = s;
    __syncthreads();
    if (t == 0) { unsigned c = 0; for (int i = 0; i < 1024; ++i) { pre[i] = c; c += part[i]; } }
    __syncthreads();
    unsigned cum = pre[t];   // count of entries in bins strictly above this chunk
    for (int b = base + 63; b >= base; --b) {
        const unsigned c2 = cum + hist[b];
        if (cum < target && c2 >= target) {
            if (phase == 0) { ctrl[0] = (unsigned)b; ctrl[1] = cum; ctrl[2] = target - cum; }
            else            { ctrl[3] = (unsigned)b; ctrl[4] = cum; ctrl[5] = target - cum; }
        }
        cum = c2;
    }
}

// -------------------------- second histogram: low 16 bits within threshold bin T
__global__ void k_hist2(const float* __restrict__ logits, const unsigned* __restrict__ ctrl,
                        unsigned* __restrict__ hist2, unsigned NN) {
    const unsigned T = ctrl[0];
    const unsigned stride = gridDim.x * blockDim.x;
    for (unsigned p = blockIdx.x * blockDim.x + threadIdx.x; p < NN; p += stride) {
        const unsigned o = ordered_of(logits[p]);
        if ((o >> 16) == T) atomicAdd(&hist2[o & 0xFFFFu], 1u);
    }
}

// ---------------------------------------------------- compact exact top-K indices
__global__ void k_compact(const float* __restrict__ logits, unsigned* ctrl,
                          unsigned* __restrict__ sel, unsigned NN) {
    const unsigned thr = (ctrl[0] << 16) | ctrl[3];   // exact 32-bit ordered threshold
    const unsigned n_above = ctrl[1] + ctrl[4];
    const unsigned rem2 = ctrl[5];
    const unsigned stride = gridDim.x * blockDim.x;
    for (unsigned p = blockIdx.x * blockDim.x + threadIdx.x; p < NN; p += stride) {
        const unsigned o = ordered_of(logits[p]);
        if (o > thr) {
            const unsigned slot = atomicAdd(&ctrl[6], 1u);
            sel[slot] = p;
        } else if (o == thr) {
            const unsigned q = atomicAdd(&ctrl[7], 1u);
            if (q < rem2) sel[n_above + q] = p;   // exact-duplicate ties only
        }
    }
}

// ------------- bitonic sort of K entries by (logit desc, idx asc) in 256KB LDS,
//               then emit edge_index (reference order) + src/tgt arrays
__global__ void k_sort_write(const float* __restrict__ logits, const unsigned* __restrict__ sel,
                             float* __restrict__ outEdge, unsigned* __restrict__ srcArr,
                             unsigned* __restrict__ tgtArr, int N, int K) {
    extern __shared__ unsigned long long keys[];   // SORTN * 8B = 256KB (<=320KB LDS)
    const int t = threadIdx.x;
    for (int i = t; i < SORTN; i += 1024) {
        if (i < K) {
            const unsigned p = sel[i];
            const unsigned o = ordered_of(logits[p]);
            keys[i] = ((unsigned long long)o << 32) | (unsigned)(~p); // tie: smaller idx first
        } else {
            keys[i] = 0ULL;    // sinks to the end under descending sort
        }
    }
    __syncthreads();
    for (unsigned k = 2; k <= SORTN; k <<= 1) {
        for (unsigned j = k >> 1; j > 0; j >>= 1) {
            for (unsigned i = t; i < SORTN; i += 1024) {
                const unsigned ixj = i ^ j;
                if (ixj > i) {
                    const unsigned long long a = keys[i], b = keys[ixj];
                    // overall DESCENDING sort
                    if (((i & k) == 0) ? (a < b) : (a > b)) { keys[i] = b; keys[ixj] = a; }
                }
            }
            __syncthreads();
        }
    }
    for (int i = t; i < K; i += 1024) {
        const unsigned p = ~(unsigned)(keys[i] & 0xFFFFFFFFu);
        const unsigned s = p / (unsigned)N, g = p % (unsigned)N;
        outEdge[i]     = (float)s;      // edge_index row 0 (src)
        outEdge[K + i] = (float)g;      // edge_index row 1 (tgt)
        srcArr[i] = s;
        tgtArr[i] = g;
    }
}

// ------------- Phase 2: probs + edge features. One wave per 16 edges; projector
//               GEMM (16x128 @ 128x32) via V_WMMA_F32_16X16X4_F32.
__global__ void k_phase2(const float* __restrict__ x, const float* __restrict__ hs,
                         const float* __restrict__ htb, const float* __restrict__ W2,
                         const float* __restrict__ b2, const float* __restrict__ Wp,
                         const float* __restrict__ bp, const unsigned* __restrict__ srcArr,
                         const unsigned* __restrict__ tgtArr, float* __restrict__ outFeat,
                         float* __restrict__ partial, int N, int D, int DE, int K) {
    __shared__ float pairT[16 * 128];   // 16 edges x 2D
    __shared__ float wpT[128 * 32];     // Wp [2D][DE]
    __shared__ float probS[16];
    const int e0 = blockIdx.x * 16;
    const int L = threadIdx.x, mr = L & 15, hi = L >> 4;

    for (int idx = L; idx < 2 * D * DE; idx += 32) wpT[idx] = Wp[idx];
    for (int e = 0; e < 16; ++e) {
        const int edge = e0 + e;
        const unsigned s = (edge < K) ? srcArr[edge] : 0u;
        const unsigned g = (edge < K) ? tgtArr[edge] : 0u;
        pairT[e * 128 + L]      = x[(size_t)s * D + L];
        pairT[e * 128 + 32 + L] = x[(size_t)s * D + 32 + L];
        pairT[e * 128 + 64 + L] = x[(size_t)g * D + L];
        pairT[e * 128 + 96 + L] = x[(size_t)g * D + 32 + L];
    }
    __syncthreads();

    // probs: lane handles edge (L%16), half-range of d given by L/16
    const int myEdge = e0 + mr;
    float acc = 0.0f;
    if (myEdge < K) {
        const unsigned s = srcArr[myEdge], g = tgtArr[myEdge];
        for (int d = hi * 32; d < hi * 32 + 32; ++d)
            acc = fmaf(gelu_f(hs[(size_t)s * D + d] + htb[(size_t)g * D + d]), W2[d], acc);
    }
    acc += __shfl_xor(acc, 16);
    const float prob = 1.0f / (1.0f + expf(-(acc + b2[0])));
    if (hi == 0) probS[mr] = prob;

    // deterministic per-block loss partial
    float contrib = (hi == 0 && myEdge < K) ? (1.0f - prob) : 0.0f;
    for (int off = 16; off >= 1; off >>= 1) contrib += __shfl_xor(contrib, off);
    if (L == 0) partial[blockIdx.x] = contrib;
    __syncthreads();

    // projector GEMM: 32 k-steps x 2 N-tiles
    v8f c0 = {}, c1 = {};
    for (int k0 = 0; k0 < 2 * D; k0 += 4) {
        const int ka = k0 + hi * 2;
        v2f a, bA, bB;
        a.x  = pairT[mr * 128 + ka];
        a.y  = pairT[mr * 128 + ka + 1];
        bA.x = wpT[ka * DE + mr];
        bA.y = wpT[(ka + 1) * DE + mr];
        bB.x = wpT[ka * DE + 16 + mr];
        bB.y = wpT[(ka + 1) * DE + 16 + mr];
        c0 = __builtin_amdgcn_wmma_f32_16x16x4_f32(false, a, false, bA, (short)0, c0,
                                                   false, false);
        c1 = __builtin_amdgcn_wmma_f32_16x16x4_f32(false, a, false, bB, (short)0, c1,
                                                   false, false);
    }
    for (int r = 0; r < 8; ++r) {
        const int M = r + hi * 8;
        const int eg = e0 + M;
        if (eg < K) {
            const float pr = probS[M];
            outFeat[(size_t)eg * DE + mr]      = gelu_f(c0[r] + bp[mr]) * pr;
            outFeat[(size_t)eg * DE + 16 + mr] = gelu_f(c1[r] + bp[16 + mr]) * pr;
        }
    }
}

// --------------------------------------- fixed-order loss reduction (deterministic)
__global__ void k_loss(const float* __restrict__ partial, int nb, float* outLoss, int K) {
    __shared__ float sm[256];
    float s = 0.0f;
    for (int i = threadIdx.x; i < nb; i += 256) s += partial[i];
    sm[threadIdx.x] = s;
    __syncthreads();
    for (int o = 128; o >= 1; o >>= 1) {
        if (threadIdx.x < o) sm[threadIdx.x] += sm[threadIdx.x + o];
        __syncthreads();
    }
    if (threadIdx.x == 0) *outLoss = sm[0] / (float)K;
}

// ---------------------------------------------------------------------------
extern "C" void kernel_launch(void* const* d_in, const int* in_sizes, int n_in,
                              void* d_out, int out_size, void* d_ws, size_t ws_size,
                              hipStream_t stream) {
    const float* x  = (const float*)d_in[0];
    const float* W1 = (const float*)d_in[1];
    const float* b1 = (const float*)d_in[2];
    const float* W2 = (const float*)d_in[3];
    const float* b2 = (const float*)d_in[4];
    const float* Wp = (const float*)d_in[5];
    const float* bp = (const float*)d_in[6];

    const int D  = in_sizes[2];                 // 64
    const int N  = in_sizes[0] / D;             // 2048
    const int DE = in_sizes[6];                 // 32
    const int K  = (out_size - 1) / (2 + DE);   // 20961
    const unsigned NN = (unsigned)N * (unsigned)N;

    // workspace layout (256B aligned), ~18.5MB total
    char* base = (char*)d_ws;
    size_t off = 0;
    auto alloc = [&](size_t bytes) { size_t o = off; off = (off + bytes + 255) & ~(size_t)255; return o; };
    float*    hs      = (float*)(base + alloc((size_t)N * D * 4));
    float*    htb     = (float*)(base + alloc((size_t)N * D * 4));
    float*    logits  = (float*)(base + alloc((size_t)NN * 4));
    unsigned* hist1   = (unsigned*)(base + alloc(HIST_BINS * 4));   // hist1/hist2/ctrl
    unsigned* hist2   = (unsigned*)(base + alloc(HIST_BINS * 4));   // contiguous for
    unsigned* ctrl    = (unsigned*)(base + alloc(256));             // one zero-pass
    unsigned* sel     = (unsigned*)(base + alloc((size_t)K * 4));
    unsigned* srcArr  = (unsigned*)(base + alloc((size_t)K * 4));
    unsigned* tgtArr  = (unsigned*)(base + alloc((size_t)K * 4));
    const int nb2     = (K + 15) / 16;
    float*    partial = (float*)(base + alloc((size_t)nb2 * 4));
    (void)ws_size; (void)n_in;

    float* outEdge = (float*)d_out;                      // [2, K] as floats
    float* outFeat = outEdge + (size_t)2 * K;            // [K, DE]
    float* outLoss = outFeat + (size_t)K * DE;           // scalar

    // 1) zero hist1+hist2+ctrl (contiguous uints)
    const unsigned nz = HIST_BINS * 2 + 64;
    k_zero<<<(nz + 255) / 256, 256, 0, stream>>>(hist1, nz);

    // 2) hs / htb via fp32 WMMA
    k_gemm_hsht<<<dim3(N / 16, D / 16, 2), 32, 0, stream>>>(x, W1, b1, hs, htb, N, D);

    // 3) all-pairs logits + 64K-bin histogram
    k_logits_hist<<<dim3(N / 64, N / 64), 256, 0, stream>>>(hs, htb, W2, b2, logits,
                                                            hist1, N, D);

    // 4) two-level radix select -> exact ordered-uint threshold
    k_scan<<<1, 1024, 0, stream>>>(hist1, ctrl, 0, (unsigned)K);
    k_hist2<<<1024, 256, 0, stream>>>(logits, ctrl, hist2, NN);
    k_scan<<<1, 1024, 0, stream>>>(hist2, ctrl, 1, (unsigned)K);

    // 5) compact exactly K pair indices
    k_compact<<<1024, 256, 0, stream>>>(logits, ctrl, sel, NN);

    // 6) bitonic sort in 256KB LDS -> reference (descending) order; emit edge_index
    k_sort_write<<<1, 1024, SORTN * sizeof(unsigned long long), stream>>>(
        logits, sel, outEdge, srcArr, tgtArr, N, K);

    // 7) phase 2: probs + WMMA edge features + loss partials
    k_phase2<<<nb2, 32, 0, stream>>>(x, hs, htb, W2, b2, Wp, bp, srcArr, tgtArr,
                                     outFeat, partial, N, D, DE, K);

    // 8) deterministic loss reduction
    k_loss<<<1, 256, 0, stream>>>(partial, nb2, outLoss, K);
}